// SSM_Transformer_627065225420
// MI455X (gfx1250) — compile-verified
//
#include <hip/hip_runtime.h>
#include <math.h>

typedef __attribute__((ext_vector_type(2))) float v2f;
typedef __attribute__((ext_vector_type(8))) float v8f;
typedef __attribute__((ext_vector_type(4))) unsigned int v4u;
typedef __attribute__((ext_vector_type(4))) int v4i;
typedef __attribute__((ext_vector_type(8))) int v8i;

#define D_MODEL 128
#define SEQ_L   2048
#define BATCH   8
#define NHEAD   4
#define HDIM    32
#define FFDIM   512
#define MROWS   (BATCH * SEQ_L)

// GEMM epilogue modes
#define GM_PLAIN  0
#define GM_POSENC 1
#define GM_RELU   2
#define GM_RES    3

// ---------------------------------------------------------------------------
// TDM: issue a 2-D TENSOR_LOAD_TO_LDS of a [tileH x tileW] fp32 tile whose
// first element is at gtile, rows strideElems apart, into LDS byte offset
// ldsOff. D# bitfields per CDNA5 ISA:
//   group0: count=1 | lds_addr | global_addr[56:0] | type=2
//   group1: data_size=2 (4B), tensor_dim0/1, tile_dim0/1, tensor_dim0_stride
// Groups 2/3 zero (2-D tensor; higher tile dims unused).
// ---------------------------------------------------------------------------
__device__ __forceinline__ void tdm_load_2d_f32(unsigned ldsOff, const float* gtile,
                                                unsigned tileW, unsigned tileH,
                                                unsigned tensorW, unsigned tensorH,
                                                unsigned strideElems)
{
    const unsigned long long ga = (unsigned long long)(size_t)gtile;
    v4u g0;
    g0.x = 1u;                                            // count=1 (valid), user mode
    g0.y = ldsOff;                                        // lds_addr (bytes)
    g0.z = (unsigned)(ga & 0xFFFFFFFFull);                // global_addr[31:0]
    g0.w = (unsigned)((ga >> 32) & 0x01FFFFFFull) | (2u << 30);  // addr[56:32] | type=2
    v8i g1;
    g1[0] = (int)(2u << 16);                              // data_size = 4 bytes
    g1[1] = (int)((tensorW & 0xFFFFu) << 16);             // tensor_dim0[15:0]
    g1[2] = (int)((tensorW >> 16) | ((tensorH & 0xFFFFu) << 16));  // dim0 hi | dim1 lo
    g1[3] = (int)((tensorH >> 16) | (tileW << 16));       // dim1 hi | tile_dim0
    g1[4] = (int)(tileH & 0xFFFFu);                       // tile_dim1 (tile_dim2=0)
    g1[5] = (int)strideElems;                             // tensor_dim0_stride[31:0]
    g1[6] = 0;                                            // stride hi | dim1_stride lo
    g1[7] = 0;
    const v4i z4 = {};
#if defined(__clang_major__) && (__clang_major__ >= 23)
    const v8i z8 = {};
    __builtin_amdgcn_tensor_load_to_lds(g0, g1, z4, z4, z8, 0);
#else
    __builtin_amdgcn_tensor_load_to_lds(g0, g1, z4, z4, 0);
#endif
}

// ---------------------------------------------------------------------------
// Generic fp32 WMMA GEMM: C[M,N] = A[M,K] @ W[K,N] + bias  (+ epilogue)
// One wave computes one 16x16 tile via V_WMMA_F32_16X16X4_F32 accumulation.
// Block = 256 threads = 8 waves stacked along M (128 rows x 16 cols / block).
// The shared W[K x 16] tile is DMA'd into LDS once by wave 0 via the TDM
// (tensor_load_to_lds + s_wait_tensorcnt), then all waves read B-fragments
// from LDS; A-fragments are contiguous 8B vector loads from global.
// ---------------------------------------------------------------------------
__global__ void gemm_wmma_f32(const float* __restrict__ A,
                              const float* __restrict__ W,
                              const float* __restrict__ bias,
                              const float* res,
                              float* C,
                              int N, int K, int mode)
{
    __shared__ float sW[FFDIM * 16];      // max K=512 -> 32KB of the 320KB WGP LDS

    const int lane = threadIdx.x & 31;
    const int wave = threadIdx.x >> 5;
    const int half = lane >> 4;           // 0: lanes 0-15, 1: lanes 16-31
    const int l16  = lane & 15;
    const int mtile = blockIdx.y * 128 + wave * 16;
    const int ntile = blockIdx.x * 16;

    if (wave == 0) {
        tdm_load_2d_f32((unsigned)(size_t)sW, W + ntile,
                        16u, (unsigned)K, (unsigned)N, (unsigned)K, (unsigned)N);
        __builtin_amdgcn_s_wait_tensorcnt(0);
    }
    __syncthreads();

    v8f acc = {};
    const float* arow = A + (unsigned)(mtile + l16) * (unsigned)K;
    for (int k = 0; k < K; k += 4) {
        const int ka = k + 2 * half;      // per-lane K pair (ISA f32 A/B layout)
        const v2f af = *(const v2f*)(arow + ka);          // contiguous, 8B aligned
        v2f bf;
        bf.x = sW[ka * 16 + l16];         // B-frag from LDS tile
        bf.y = sW[(ka + 1) * 16 + l16];
        acc = __builtin_amdgcn_wmma_f32_16x16x4_f32(false, af, false, bf,
                                                    (short)0, acc, false, false);
    }

    const int col = ntile + l16;
    const float bv = bias[col];
#pragma unroll
    for (int r = 0; r < 8; ++r) {
        const unsigned row = (unsigned)(mtile + r + 8 * half);  // ISA 16x16 C/D layout
        float v = acc[r] + bv;
        if (mode == GM_POSENC) {
            const int l = (int)(row & (SEQ_L - 1));
            // pe[l,c]: i=c/2; f=exp(-(2i)*ln(10000)/D); even->sin, odd->cos
            const float freq = expf(-0.07195583f * (float)(col & ~1));
            const float ang  = (float)l * freq;
            v += (col & 1) ? cosf(ang) : sinf(ang);
        } else if (mode == GM_RELU) {
            v = fmaxf(v, 0.0f);
        } else if (mode == GM_RES) {
            v += res[row * (unsigned)N + (unsigned)col];
        }
        C[row * (unsigned)N + (unsigned)col] = v;
    }
}

// ---------------------------------------------------------------------------
// SSM recurrence: h_t = tanh(A*h_{t-1} + Bx_t), in place on Bx. Strictly
// sequential in t; parallel over (batch, channel) = 1024 lanes spread as
// 32 single-wave blocks so many WGPs hide the tanh latency chain.
// ---------------------------------------------------------------------------
__global__ void ssm_scan(float* bx, const float* __restrict__ Av)
{
    const int b = blockIdx.x;
    const int d = blockIdx.y * 32 + threadIdx.x;
    const float a = Av[d];
    float* p = bx + (unsigned)b * (SEQ_L * D_MODEL) + d;
    float h = 0.0f;
    for (int t = 0; t < SEQ_L; ++t) {
        const float v = p[(unsigned)t * D_MODEL];
        h = tanhf(fmaf(a, h, v));
        p[(unsigned)t * D_MODEL] = h;
    }
}

// ---------------------------------------------------------------------------
// LayerNorm over D=128 per row. One wave32 per row, 8 rows per block.
// ---------------------------------------------------------------------------
__global__ void layernorm_rows(const float* __restrict__ x,
                               const float* __restrict__ gam,
                               const float* __restrict__ bet,
                               float* __restrict__ y)
{
    const int lane = threadIdx.x & 31;
    const int wave = threadIdx.x >> 5;
    const unsigned row = (unsigned)(blockIdx.x * 8 + wave);
    const float* xr = x + row * D_MODEL;
    const float v0 = xr[lane], v1 = xr[lane + 32], v2 = xr[lane + 64], v3 = xr[lane + 96];
    float s = v0 + v1 + v2 + v3;
    float q = v0 * v0 + v1 * v1 + v2 * v2 + v3 * v3;
    for (int off = 16; off; off >>= 1) {
        s += __shfl_xor(s, off, 32);
        q += __shfl_xor(q, off, 32);
    }
    const float mu   = s * (1.0f / 128.0f);
    const float var  = q * (1.0f / 128.0f) - mu * mu;
    const float rstd = rsqrtf(var + 1e-5f);
    float* yr = y + row * D_MODEL;
    yr[lane]      = (v0 - mu) * rstd * gam[lane]      + bet[lane];
    yr[lane + 32] = (v1 - mu) * rstd * gam[lane + 32] + bet[lane + 32];
    yr[lane + 64] = (v2 - mu) * rstd * gam[lane + 64] + bet[lane + 64];
    yr[lane + 96] = (v3 - mu) * rstd * gam[lane + 96] + bet[lane + 96];
}

// ---------------------------------------------------------------------------
// Attention for one (batch, head, 16-row q-tile). qkv layout: [B,L,3*D],
// head h columns: Q at h*32, K at 128+h*32, V at 256+h*32.
// Full 16x2048 score tile lives in LDS (128KB of the 320KB WGP LDS).
// Phase 1: scores = Q K^T / sqrt(32) via WMMA; Q-frags hoisted to registers
// Phase 2: in-LDS softmax, one wave per 2 rows, wave32 shuffle reductions
// Phase 3: O = P V via WMMA (P-frags are contiguous LDS v2f reads),
//          cross-wave reduce with ds_add_f32 atomics
// ---------------------------------------------------------------------------
#define SSTR 2052  // padded row stride to dodge LDS bank conflicts

__global__ void attn_wmma(const float* __restrict__ qkv, float* __restrict__ o)
{
    extern __shared__ float smem[];
    float* sc  = smem;                // [16][SSTR] scores
    float* osc = smem + 16 * SSTR;    // [16][32] output accumulator

    const int qt   = blockIdx.x;      // 0..127
    const int h    = blockIdx.y;      // 0..3
    const int b    = blockIdx.z;      // 0..7
    const int lane = threadIdx.x & 31;
    const int wave = threadIdx.x >> 5;
    const int half = lane >> 4;
    const int l16  = lane & 15;
    const int qbase = qt * 16;

    const unsigned rs = 3 * D_MODEL;  // 384
    const float* base = qkv + (unsigned)b * (SEQ_L * 3 * D_MODEL);
    const float* qp = base + h * HDIM;
    const float* kp = base + D_MODEL + h * HDIM;
    const float* vp = base + 2 * D_MODEL + h * HDIM;

    // ---- Phase 1: score tiles -------------------------------------------
    // Q fragments are invariant across all 16 key tiles: hoist into regs.
    v2f qa[HDIM / 4];
#pragma unroll
    for (int kk8 = 0; kk8 < HDIM / 4; ++kk8) {
        const int ka = kk8 * 4 + 2 * half;
        qa[kk8] = *(const v2f*)(qp + (unsigned)(qbase + l16) * rs + ka);
    }
    const float scale = 0.17677669529663687f;  // 1/sqrt(32)
    for (int j = 0; j < 16; ++j) {
        const int kt = wave * 256 + j * 16;
        v8f acc = {};
#pragma unroll
        for (int kk8 = 0; kk8 < HDIM / 4; ++kk8) {
            const int ka = kk8 * 4 + 2 * half;
            const v2f bf = *(const v2f*)(kp + (unsigned)(kt + l16) * rs + ka);  // B = K^T
            acc = __builtin_amdgcn_wmma_f32_16x16x4_f32(false, qa[kk8], false, bf,
                                                        (short)0, acc, false, false);
        }
#pragma unroll
        for (int r = 0; r < 8; ++r)
            sc[(r + 8 * half) * SSTR + kt + l16] = acc[r] * scale;
    }
    __syncthreads();

    // ---- Phase 2: softmax (wave w handles rows 2w, 2w+1) ----------------
    for (int rr = 0; rr < 2; ++rr) {
        float* row = sc + (wave * 2 + rr) * SSTR;
        float m = -3.4e38f;
        for (int c = lane; c < SEQ_L; c += 32) m = fmaxf(m, row[c]);
        for (int off = 16; off; off >>= 1) m = fmaxf(m, __shfl_xor(m, off, 32));
        float s = 0.0f;
        for (int c = lane; c < SEQ_L; c += 32) {
            const float e = expf(row[c] - m);
            row[c] = e;
            s += e;
        }
        for (int off = 16; off; off >>= 1) s += __shfl_xor(s, off, 32);
        const float inv = 1.0f / s;
        for (int c = lane; c < SEQ_L; c += 32) row[c] *= inv;
    }
    __syncthreads();
    for (int i = threadIdx.x; i < 512; i += 256) osc[i] = 0.0f;
    __syncthreads();

    // ---- Phase 3: O = P @ V (wave owns key span [w*256, w*256+256)) -----
    for (int nt = 0; nt < 2; ++nt) {
        v8f acc = {};
        const int k0beg = wave * 256;
        for (int k0 = k0beg; k0 < k0beg + 256; k0 += 4) {
            const int ka = k0 + 2 * half;
            const v2f af = *(const v2f*)(sc + l16 * SSTR + ka);  // contiguous LDS read
            v2f bf;
            bf.x = vp[(unsigned)ka * rs + nt * 16 + l16];
            bf.y = vp[(unsigned)(ka + 1) * rs + nt * 16 + l16];
            acc = __builtin_amdgcn_wmma_f32_16x16x4_f32(false, af, false, bf,
                                                        (short)0, acc, false, false);
        }
#pragma unroll
        for (int r = 0; r < 8; ++r)
            atomicAdd(&osc[(r + 8 * half) * 32 + nt * 16 + l16], acc[r]);  // ds_add_f32
    }
    __syncthreads();
    for (int i = threadIdx.x; i < 512; i += 256) {
        const int row = i >> 5, col = i & 31;
        o[(unsigned)(b * SEQ_L + qbase + row) * D_MODEL + (unsigned)(h * HDIM + col)] = osc[i];
    }
}

// ---------------------------------------------------------------------------
// Head: LN(last token) -> l1 + exact GELU -> l2. One block (128 thr) / batch.
// ---------------------------------------------------------------------------
__global__ void head_kernel(const float* __restrict__ hbuf,
                            const float* __restrict__ lns, const float* __restrict__ lnb,
                            const float* __restrict__ l1w, const float* __restrict__ l1b,
                            const float* __restrict__ l2w, const float* __restrict__ l2b,
                            float* __restrict__ out)
{
    __shared__ float zsh[128];
    __shared__ float reds[4], redq[4], redo[4];
    const int d = threadIdx.x;
    const int lane = d & 31, wave = d >> 5;
    const int b = blockIdx.x;

    const float xv = hbuf[(unsigned)(b * SEQ_L + SEQ_L - 1) * D_MODEL + d];
    float s = xv, q = xv * xv;
    for (int off = 16; off; off >>= 1) {
        s += __shfl_xor(s, off, 32);
        q += __shfl_xor(q, off, 32);
    }
    if (lane == 0) { reds[wave] = s; redq[wave] = q; }
    __syncthreads();
    s = reds[0] + reds[1] + reds[2] + reds[3];
    q = redq[0] + redq[1] + redq[2] + redq[3];
    const float mu  = s * (1.0f / 128.0f);
    const float var = q * (1.0f / 128.0f) - mu * mu;
    const float z = (xv - mu) * rsqrtf(var + 1e-5f) * lns[d] + lnb[d];
    zsh[d] = z;
    __syncthreads();

    float acc = l1b[d];
    for (int k = 0; k < 128; ++k) acc = fmaf(zsh[k], l1w[k * 128 + d], acc);
    const float g = 0.5f * acc * (1.0f + erff(acc * 0.70710678118f));  // exact GELU
    float part = g * l2w[d];
    for (int off = 16; off; off >>= 1) part += __shfl_xor(part, off, 32);
    if (lane == 0) redo[wave] = part;
    __syncthreads();
    if (d == 0) out[b] = redo[0] + redo[1] + redo[2] + redo[3] + l2b[0];
}

// ---------------------------------------------------------------------------
extern "C" void kernel_launch(void* const* d_in, const int* in_sizes, int n_in,
                              void* d_out, int out_size, void* d_ws, size_t ws_size,
                              hipStream_t stream)
{
    (void)out_size; (void)ws_size; (void)n_in;

    // Detect pytree flattening order: x (1,048,576 elems) is unique.
    const bool insertion = (in_sizes[0] == BATCH * SEQ_L * 64);
    int xi, ipw, ipb, hls, hlb, h1w, h1b, h2w, h2b;
    int sA[2], sBw[2], sBb[2], sCw[2], sCb[2];
    int tQw[2], tQb[2], tOw[2], tOb[2], t1s[2], t1b[2], t2s[2], t2b[2];
    int tF1w[2], tF1b[2], tF2w[2], tF2b[2];
    if (insertion) {
        xi = 0; ipw = 1; ipb = 2;
        for (int i = 0; i < 2; ++i) { int o = 3 + 5 * i;
            sA[i]=o; sBw[i]=o+1; sBb[i]=o+2; sCw[i]=o+3; sCb[i]=o+4; }
        for (int i = 0; i < 2; ++i) { int o = 13 + 12 * i;
            tQw[i]=o; tQb[i]=o+1; tOw[i]=o+2; tOb[i]=o+3;
            t1s[i]=o+4; t1b[i]=o+5; t2s[i]=o+6; t2b[i]=o+7;
            tF1w[i]=o+8; tF1b[i]=o+9; tF2w[i]=o+10; tF2b[i]=o+11; }
        hls = 37; hlb = 38; h1w = 39; h1b = 40; h2w = 41; h2b = 42;
    } else {  // JAX sorted-key order: params before x, dict keys sorted
        h1b = 0; h1w = 1; h2b = 2; h2w = 3; hlb = 4; hls = 5;
        ipb = 6; ipw = 7;
        for (int i = 0; i < 2; ++i) { int o = 8 + 5 * i;
            sA[i]=o; sBb[i]=o+1; sBw[i]=o+2; sCb[i]=o+3; sCw[i]=o+4; }
        for (int i = 0; i < 2; ++i) { int o = 18 + 12 * i;
            tF1b[i]=o; tF1w[i]=o+1; tF2b[i]=o+2; tF2w[i]=o+3;
            t1b[i]=o+4; t1s[i]=o+5; t2b[i]=o+6; t2s[i]=o+7;
            tOb[i]=o+8; tOw[i]=o+9; tQb[i]=o+10; tQw[i]=o+11; }
        xi = 42;
    }
    auto F = [&](int i) { return (const float*)d_in[i]; };

    float* buf0 = (float*)d_ws;                           // h        [16384,128]
    float* buf1 = buf0 + (size_t)MROWS * D_MODEL;         // qkv/ff   [16384,512]
    float* buf2 = buf1 + (size_t)MROWS * FFDIM;           // scratch  [16384,128]

    const dim3 blk(256);
    const size_t attn_smem = (size_t)(16 * SSTR + 512) * sizeof(float);  // ~130 KB

    // h = x @ in_proj + b + posenc
    gemm_wmma_f32<<<dim3(D_MODEL / 16, MROWS / 128), blk, 0, stream>>>(
        F(xi), F(ipw), F(ipb), nullptr, buf0, D_MODEL, 64, GM_POSENC);

    // SSM layers
    for (int i = 0; i < 2; ++i) {
        gemm_wmma_f32<<<dim3(8, 128), blk, 0, stream>>>(
            buf0, F(sBw[i]), F(sBb[i]), nullptr, buf2, 128, 128, GM_PLAIN);
        ssm_scan<<<dim3(8, 4), 32, 0, stream>>>(buf2, F(sA[i]));
        gemm_wmma_f32<<<dim3(8, 128), blk, 0, stream>>>(
            buf2, F(sCw[i]), F(sCb[i]), nullptr, buf0, 128, 128, GM_PLAIN);
    }

    // Transformer encoder layers (pre-norm)
    for (int i = 0; i < 2; ++i) {
        layernorm_rows<<<2048, blk, 0, stream>>>(buf0, F(t1s[i]), F(t1b[i]), buf2);
        gemm_wmma_f32<<<dim3(24, 128), blk, 0, stream>>>(
            buf2, F(tQw[i]), F(tQb[i]), nullptr, buf1, 384, 128, GM_PLAIN);
        attn_wmma<<<dim3(128, 4, 8), blk, attn_smem, stream>>>(buf1, buf2);
        gemm_wmma_f32<<<dim3(8, 128), blk, 0, stream>>>(
            buf2, F(tOw[i]), F(tOb[i]), buf0, buf0, 128, 128, GM_RES);
        layernorm_rows<<<2048, blk, 0, stream>>>(buf0, F(t2s[i]), F(t2b[i]), buf2);
        gemm_wmma_f32<<<dim3(32, 128), blk, 0, stream>>>(
            buf2, F(tF1w[i]), F(tF1b[i]), nullptr, buf1, 512, 128, GM_RELU);
        gemm_wmma_f32<<<dim3(8, 128), blk, 0, stream>>>(
            buf1, F(tF2w[i]), F(tF2b[i]), buf0, buf0, 128, 512, GM_RES);
    }

    head_kernel<<<8, 128, 0, stream>>>(buf0, F(hls), F(hlb), F(h1w), F(h1b),
                                       F(h2w), F(h2b), (float*)d_out);
}